// GraphAutoencoder_69097433858683
// MI455X (gfx1250) — compile-verified
//
#include <hip/hip_runtime.h>

typedef __attribute__((ext_vector_type(2))) float v2f;
typedef __attribute__((ext_vector_type(8))) float v8f;

#define N_NODES 50000

// ---------------------------------------------------------------------------
// Utility kernels
// ---------------------------------------------------------------------------
__global__ void zero_i32_kernel(int* __restrict__ p, int n) {
    int i = blockIdx.x * blockDim.x + threadIdx.x;
    if (i < n) p[i] = 0;
}

__global__ void zero_f32_kernel(float* __restrict__ p, long long n) {
    long long i = (long long)blockIdx.x * blockDim.x + threadIdx.x;
    if (i < n) p[i] = 0.0f;
}

// Count in/out degrees (shared by all three layers: same src/dst).
__global__ void count_deg_kernel(const int* __restrict__ src, const int* __restrict__ dst,
                                 int* __restrict__ degO, int* __restrict__ degI, int E) {
    int i = blockIdx.x * blockDim.x + threadIdx.x;
    if (i < E) {
        atomicAdd(&degO[src[i]], 1);
        atomicAdd(&degI[dst[i]], 1);
    }
}

// rout = rsqrt(max(deg_out,1)), rin = rsqrt(max(deg_in,1))
__global__ void deg_rsqrt_kernel(const int* __restrict__ degO, const int* __restrict__ degI,
                                 float* __restrict__ rout, float* __restrict__ rin, int n) {
    int i = blockIdx.x * blockDim.x + threadIdx.x;
    if (i < n) {
        int dO = degO[i]; if (dO < 1) dO = 1;
        int dI = degI[i]; if (dI < 1) dI = 1;
        rout[i] = rsqrtf((float)dO);
        rin[i]  = rsqrtf((float)dI);
    }
}

// ---------------------------------------------------------------------------
// Fused row-scaled GEMM with fp32 WMMA:  C = diag(rowscale) * A @ W (+ bias)
// A: [M,K] row-major, W: [K,N] row-major, C: [M,N].
// M,K,N all multiples of 16 (M=50000=3125*16, K/N in {32,64,128}).
// One wave per 16x16 C tile; K stepped by 4 via V_WMMA_F32_16X16X4_F32.
// Branches are wave-uniform (blockDim.x==32) so EXEC is all-ones at WMMA.
// ---------------------------------------------------------------------------
__global__ __launch_bounds__(128)
void gemm_rowscale_wmma_kernel(const float* __restrict__ A,
                               const float* __restrict__ rowscale, // nullable
                               const float* __restrict__ W,
                               const float* __restrict__ bias,     // nullable
                               float* __restrict__ C,
                               int M, int K, int N) {
    const int lane   = threadIdx.x;                          // 0..31
    const int waveId = blockIdx.x * blockDim.y + threadIdx.y;
    const int tilesN = N >> 4;
    const int tm = waveId / tilesN;
    const int tn = waveId - tm * tilesN;
    if (tm >= (M >> 4)) return;                              // wave-uniform

    const int r  = lane & 15;            // row within A-tile / col within C-tile
    const int kh = (lane >> 4) << 1;     // K sub-offset: 0 or 2
    const int am = (tm << 4) + r;        // A row this lane loads
    const int cn = (tn << 4) + r;        // C column this lane owns

    const float s = rowscale ? rowscale[am] : 1.0f;
    const float* __restrict__ Arow = A + (size_t)am * K;

    v8f acc = {};
    for (int k0 = 0; k0 < K; k0 += 4) {
        v2f a, b;
        // A 16x4 fragment: lane r holds (row=am, K=k0+kh, k0+kh+1)
        a.x = Arow[k0 + kh]     * s;
        a.y = Arow[k0 + kh + 1] * s;
        // B 4x16 fragment: lane r holds (col=cn, K=k0+kh, k0+kh+1)
        b.x = W[(size_t)(k0 + kh)     * N + cn];
        b.y = W[(size_t)(k0 + kh + 1) * N + cn];
        acc = __builtin_amdgcn_wmma_f32_16x16x4_f32(
            /*neg_a=*/false, a, /*neg_b=*/false, b,
            /*c_mod=*/(short)0, acc, /*reuse_a=*/false, /*reuse_b=*/false);
    }

    const float bv = bias ? bias[cn] : 0.0f;
    // C/D layout: VGPR v, lane -> row = 16*tm + v + 8*(lane>>4), col = cn
    const int rbase = (tm << 4) + ((lane >> 4) << 3);
#pragma unroll
    for (int v = 0; v < 8; ++v) {
        C[(size_t)(rbase + v) * N + cn] = acc[v] + bv;
    }
}

// ---------------------------------------------------------------------------
// SpMM scatter: AGG[dst[e], f] += T[src[e], f] * (gscale ? gscale[src[e]] : 1)
// One thread per (edge, feature); F is a power of two (32 or 64).
// ---------------------------------------------------------------------------
__global__ void spmm_scatter_kernel(const float* __restrict__ T,
                                    const float* __restrict__ gscale, // nullable
                                    const int* __restrict__ src,
                                    const int* __restrict__ dst,
                                    float* __restrict__ AGG,
                                    int E, int F_log2) {
    long long idx = (long long)blockIdx.x * blockDim.x + threadIdx.x;
    long long total = (long long)E << F_log2;
    if (idx >= total) return;
    const int e = (int)(idx >> F_log2);
    const int f = (int)(idx & ((1 << F_log2) - 1));
    const int s = src[e];
    const int d = dst[e];
    float v = T[((size_t)s << F_log2) + f];
    if (gscale) v *= gscale[s];
    atomicAdd(&AGG[((size_t)d << F_log2) + f], v);
}

// ---------------------------------------------------------------------------
// Pointwise: out[i] = act(in[i] * rin[row] + bias[col])
// ---------------------------------------------------------------------------
__global__ void scale_bias_act_kernel(const float* __restrict__ in,
                                      const float* __restrict__ rin,
                                      const float* __restrict__ bias,
                                      float* __restrict__ out,
                                      long long total, int F_log2, int do_relu) {
    long long i = (long long)blockIdx.x * blockDim.x + threadIdx.x;
    if (i >= total) return;
    const int row = (int)(i >> F_log2);
    const int col = (int)(i & ((1 << F_log2) - 1));
    float v = in[i] * rin[row] + bias[col];
    if (do_relu) v = fmaxf(v, 0.0f);
    out[i] = v;
}

// ---------------------------------------------------------------------------
// Launch
// ---------------------------------------------------------------------------
static inline int cdiv(long long a, int b) { return (int)((a + b - 1) / b); }

extern "C" void kernel_launch(void* const* d_in, const int* in_sizes, int n_in,
                              void* d_out, int out_size, void* d_ws, size_t ws_size,
                              hipStream_t stream) {
    const float* features = (const float*)d_in[0];   // [N,128]
    const int*   src      = (const int*)  d_in[1];   // [E]
    const int*   dst      = (const int*)  d_in[2];   // [E]
    const float* W1       = (const float*)d_in[3];   // [128,64]
    const float* b1       = (const float*)d_in[4];   // [64]
    const float* W2       = (const float*)d_in[5];   // [64,32]
    const float* b2       = (const float*)d_in[6];   // [32]
    const float* W3       = (const float*)d_in[7];   // [32,128]
    const float* b3       = (const float*)d_in[8];   // [128]

    const int N = in_sizes[0] / 128;   // 50000
    const int E = in_sizes[1];

    // Workspace layout (26.4 MB):
    int*   degO = (int*)d_ws;
    int*   degI = degO + N;
    float* rout = (float*)(degI + N);
    float* rin  = rout + N;
    float* bufA = rin + N;                    // N*64 floats (T1 / T2+AGG2 / AGG3)
    float* bufB = bufA + (size_t)N * 64;      // N*64 floats (AGG1 -> H)

    float* z_out = (float*)d_out;             // [N,32]
    float* recon = z_out + (size_t)N * 32;    // [N,128]

    const int TPB = 256;

    // ---- Degrees & norms (shared across layers) ----
    zero_i32_kernel<<<cdiv(2LL * N, TPB), TPB, 0, stream>>>(degO, 2 * N);
    count_deg_kernel<<<cdiv(E, TPB), TPB, 0, stream>>>(src, dst, degO, degI, E);
    deg_rsqrt_kernel<<<cdiv(N, TPB), TPB, 0, stream>>>(degO, degI, rout, rin, N);

    const dim3 gblk(32, 4);  // 4 waves/block, 1 wave per 16x16 tile
    const int tilesM = N / 16;  // 3125

    // ---- Layer 1: 128 -> 64, transform-first ----
    // T1 = diag(rout) * X @ W1
    gemm_rowscale_wmma_kernel<<<cdiv((long long)tilesM * (64 / 16), 4), gblk, 0, stream>>>(
        features, rout, W1, nullptr, bufA, N, 128, 64);
    zero_f32_kernel<<<cdiv((long long)N * 64, TPB), TPB, 0, stream>>>(bufB, (long long)N * 64);
    spmm_scatter_kernel<<<cdiv((long long)E << 6, TPB), TPB, 0, stream>>>(
        bufA, nullptr, src, dst, bufB, E, 6);
    // H = relu(AGG1 * rin + b1) (in place)
    scale_bias_act_kernel<<<cdiv((long long)N * 64, TPB), TPB, 0, stream>>>(
        bufB, rin, b1, bufB, (long long)N * 64, 6, 1);

    // ---- Layer 2: 64 -> 32, transform-first ----
    float* T2   = bufA;                      // [N,32]
    float* AGG2 = bufA + (size_t)N * 32;     // [N,32]
    gemm_rowscale_wmma_kernel<<<cdiv((long long)tilesM * (32 / 16), 4), gblk, 0, stream>>>(
        bufB, rout, W2, nullptr, T2, N, 64, 32);
    zero_f32_kernel<<<cdiv((long long)N * 32, TPB), TPB, 0, stream>>>(AGG2, (long long)N * 32);
    spmm_scatter_kernel<<<cdiv((long long)E << 5, TPB), TPB, 0, stream>>>(
        T2, nullptr, src, dst, AGG2, E, 5);
    // z = AGG2 * rin + b2  -> directly into d_out
    scale_bias_act_kernel<<<cdiv((long long)N * 32, TPB), TPB, 0, stream>>>(
        AGG2, rin, b2, z_out, (long long)N * 32, 5, 0);

    // ---- Layer 3: 32 -> 128, aggregate-first (rout fused into gather) ----
    float* AGG3 = bufA;                      // reuse [N,32]
    zero_f32_kernel<<<cdiv((long long)N * 32, TPB), TPB, 0, stream>>>(AGG3, (long long)N * 32);
    spmm_scatter_kernel<<<cdiv((long long)E << 5, TPB), TPB, 0, stream>>>(
        z_out, rout, src, dst, AGG3, E, 5);
    // recon = diag(rin) * AGG3 @ W3 + b3
    gemm_rowscale_wmma_kernel<<<cdiv((long long)tilesM * (128 / 16), 4), gblk, 0, stream>>>(
        AGG3, rin, W3, b3, recon, N, 32, 128);
}